// GraphConvolutionSparse_82351702933664
// MI455X (gfx1250) — compile-verified
//
#include <hip/hip_runtime.h>
#include <hip/hip_bf16.h>

// GraphConvolutionSparse on MI455X (gfx1250, wave32).
//
// Pipeline (all on `stream`, graph-capture safe):
//   1) zero xw (in d_ws) and d_out (single fused kernel)
//   2) spmm_scatter: xw  += x_vals * W[x_cols]      scattered to x_rows   (1.0M nnz)
//   3) spmm_scatter: out += adj_vals * xw[adj_cols] scattered to adj_rows (1.6M nnz)
//   4) relu in place on d_out
//
// Roofline: ~0.67 GFLOP, traffic-dominated. W/xw/out (~54MB) are L2-resident
// (192MB L2), so the random 512B row gathers and scatter-atomics run at L2
// speed; HBM sees only the streamed COO arrays + output (~57MB -> ~2.5us floor
// at 23.3TB/s). WMMA deliberately unused: COO columns are uncorrelated, so a
// 16x16xK tile is >=94% structural zeros (worse bytes/FLOP than the VALU axpy)
// and densifying 16 rows (16x5000 f32 = 312.5KB LDS) would turn 0.26 GFLOP of
// useful work into 64 GFLOP over 99.6% zeros. The CDNA5 feature that fits this
// workload is the TDM.
//
// CDNA5-specific paths (verified in histogram: 6x tensor_load_to_lds,
// s_wait_tensorcnt, global_prefetch_b8, single-instruction atomics):
//   - TDM double-buffered streaming of COO metadata into LDS, overlapped with
//     gather/atomic processing; frees the VMEM pipe for the random gathers and
//     turns per-entry metadata access into free same-address LDS broadcasts.
//   - global_prefetch_b8 on the 512B gather rows.
//   - no-return global_atomic_add_f32 (relaxed, agent scope) for scatter-add.

#define GCN_N_NODES 50000
#define GCN_OUT_DIM 128   // 32 lanes x float4 per row
#define GCN_CHUNK   256   // COO entries staged per TDM chunk (8 waves x 32)

typedef unsigned int gcn_v4u __attribute__((ext_vector_type(4)));
typedef int          gcn_v8i __attribute__((ext_vector_type(8)));
typedef int          gcn_v4i __attribute__((ext_vector_type(4)));

__device__ __forceinline__ void gcn_atomic_add_f32(float* p, float v) {
  // Return value discarded -> backend uses the no-return
  // global_atomic_add_f32 (STOREcnt) form, resolved in L2.
  (void)__hip_atomic_fetch_add(p, v, __ATOMIC_RELAXED, __HIP_MEMORY_SCOPE_AGENT);
}

// Issue a 1-D TDM copy of `tile_elems` 4-byte elements from `gptr` into LDS at
// byte offset `lds_off`. `remaining` bounds tensor_dim0 so out-of-range tail
// reads return zeros (ISA: OOB loads return 0), making the tail self-masking.
__device__ __forceinline__ void gcn_tdm_load_1d(unsigned int lds_off,
                                                const void* gptr,
                                                int tile_elems,
                                                long long remaining) {
  unsigned long long ga = (unsigned long long)(uintptr_t)gptr;
  unsigned int td0 = (remaining > 0x7FFFFFFFLL) ? 0x7FFFFFFFu
                                                : (unsigned int)remaining;
  // D# group 0: count=1 (valid user desc), lds_addr, global_addr[56:0], type=2
  gcn_v4u g0;
  g0.x = 1u;
  g0.y = lds_off;
  g0.z = (unsigned int)(ga & 0xFFFFFFFFull);
  g0.w = (unsigned int)((ga >> 32) & 0x1FFFFFFull) | (2u << 30);
  // D# group 1: data_size=4B; tensor_dim0=td0; tensor_dim1=1; tile_dim0=N;
  // tile_dim1=1; tile_dim2=0; tensor_dim0_stride=td0.
  gcn_v8i g1;
  g1[0] = (int)(2u << 16);                                   // data_size=4B
  g1[1] = (int)((td0 & 0xFFFFu) << 16);                      // tdim0[15:0]
  g1[2] = (int)(((td0 >> 16) & 0xFFFFu) | (1u << 16));       // tdim0[31:16], tdim1=1
  g1[3] = (int)((unsigned int)tile_elems << 16);             // tile_dim0
  g1[4] = 1;                                                 // tile_dim1=1
  g1[5] = (int)td0;                                          // dim0_stride lo
  g1[6] = 0;
  g1[7] = 0;
  gcn_v4i g2 = {1, 0, 0, 0};                 // tensor_dim2=1 (single plane)
  gcn_v4i g3 = {0, (int)(1u << 16), 0, 0};   // tensor_dim4=1
#if __has_include(<hip/amd_detail/amd_gfx1250_TDM.h>)
  gcn_v8i gz = {0, 0, 0, 0, 0, 0, 0, 0};
  __builtin_amdgcn_tensor_load_to_lds(g0, g1, g2, g3, gz, 0);   // clang-23 form
#else
  __builtin_amdgcn_tensor_load_to_lds(g0, g1, g2, g3, 0);       // ROCm 7.2 form
#endif
}

// Zero both accumulators in one launch.
__global__ __launch_bounds__(256) void gcn_zero2_f4(float4* __restrict__ a,
                                                    float4* __restrict__ b,
                                                    int n4) {
  const float4 z = make_float4(0.f, 0.f, 0.f, 0.f);
  int i = blockIdx.x * blockDim.x + threadIdx.x;
  const int stride = gridDim.x * blockDim.x;
  for (; i < n4; i += stride) {
    a[i] = z;
    b[i] = z;
  }
}

// Block-strided over 256-entry COO chunks. Wave 0 drives the TDM double
// buffer; all 8 waves consume metadata from LDS (same-address broadcast reads)
// and do the 128-dim axpy: one global_load_b128 + 4 global_atomic_add_f32
// per lane per entry.
__global__ __launch_bounds__(256) void gcn_spmm_scatter(
    const int*   __restrict__ rows,
    const int*   __restrict__ cols,
    const float* __restrict__ vals,
    const float* __restrict__ src,   // [src_rows, 128] gather source (W or xw)
    float*       __restrict__ dst,   // [N_NODES, 128] scatter-add destination
    int nnz) {
  __shared__ int   bufR[2][GCN_CHUNK];
  __shared__ int   bufC[2][GCN_CHUNK];
  __shared__ float bufV[2][GCN_CHUNK];

  const int lane = threadIdx.x & 31;
  const int wav  = threadIdx.x >> 5;
  const int chunks = (nnz + GCN_CHUNK - 1) / GCN_CHUNK;

  int ci = blockIdx.x;
  if (ci >= chunks) return;

  // Prologue: stream the first chunk into buffer 0.
  if (threadIdx.x < 32) {
    const long long b = (long long)ci * GCN_CHUNK;
    gcn_tdm_load_1d((unsigned int)(uintptr_t)&bufR[0][0], rows + b, GCN_CHUNK, nnz - b);
    gcn_tdm_load_1d((unsigned int)(uintptr_t)&bufC[0][0], cols + b, GCN_CHUNK, nnz - b);
    gcn_tdm_load_1d((unsigned int)(uintptr_t)&bufV[0][0], vals + b, GCN_CHUNK, nnz - b);
  }

  int p = 0;
  for (; ci < chunks; ci += gridDim.x, p ^= 1) {
    if (threadIdx.x < 32) __builtin_amdgcn_s_wait_tensorcnt(0);  // chunk ci ready
    __syncthreads();   // release consumers; also proves buf[p^1] fully consumed

    // Issue-ahead: stream the next chunk while this one is processed.
    const int nci = ci + gridDim.x;
    if (nci < chunks && threadIdx.x < 32) {
      const long long b = (long long)nci * GCN_CHUNK;
      gcn_tdm_load_1d((unsigned int)(uintptr_t)&bufR[p ^ 1][0], rows + b, GCN_CHUNK, nnz - b);
      gcn_tdm_load_1d((unsigned int)(uintptr_t)&bufC[p ^ 1][0], cols + b, GCN_CHUNK, nnz - b);
      gcn_tdm_load_1d((unsigned int)(uintptr_t)&bufV[p ^ 1][0], vals + b, GCN_CHUNK, nnz - b);
    }

    int count = nnz - ci * GCN_CHUNK;
    if (count > GCN_CHUNK) count = GCN_CHUNK;
    const int kbeg = wav * 32;
    const int kend = (kbeg + 32 < count) ? (kbeg + 32) : count;

    if (kbeg < count) {
      // Prefetch this wave's 32 gather rows: 512B row = 4 x 128B lines.
      // Lowers to global_prefetch_b8 (one address per active lane).
      const int ee  = kbeg + lane;
      const int cpf = (ee < count) ? bufC[p][ee] : 0;
      const char* pf = (const char*)(src + (size_t)cpf * GCN_OUT_DIM);
      __builtin_prefetch(pf,       0, 1);
      __builtin_prefetch(pf + 128, 0, 1);
      __builtin_prefetch(pf + 256, 0, 1);
      __builtin_prefetch(pf + 384, 0, 1);

      // Unroll x2 so the next entry's b128 gather issues under the current
      // entry's four atomics (software pipelining across the L2 latency).
#pragma unroll 2
      for (int k = kbeg; k < kend; ++k) {
        const int   ck = bufC[p][k];   // same-address LDS reads: broadcast
        const int   rk = bufR[p][k];
        const float vk = bufV[p][k];
        // Coalesced: lane l covers bytes [16l, 16l+16) of the 512B row.
        const float4 w = ((const float4*)(src + (size_t)ck * GCN_OUT_DIM))[lane];
        float* o = dst + (size_t)rk * GCN_OUT_DIM + lane * 4;
        gcn_atomic_add_f32(o + 0, vk * w.x);
        gcn_atomic_add_f32(o + 1, vk * w.y);
        gcn_atomic_add_f32(o + 2, vk * w.z);
        gcn_atomic_add_f32(o + 3, vk * w.w);
      }
    }
    __syncthreads();   // all done with buf[p] before it is refilled next round
  }
}

__global__ __launch_bounds__(256) void gcn_relu_f4(float4* __restrict__ a, int n4) {
  int i = blockIdx.x * blockDim.x + threadIdx.x;
  const int stride = gridDim.x * blockDim.x;
  for (; i < n4; i += stride) {
    float4 t = a[i];
    t.x = fmaxf(t.x, 0.f);
    t.y = fmaxf(t.y, 0.f);
    t.z = fmaxf(t.z, 0.f);
    t.w = fmaxf(t.w, 0.f);
    a[i] = t;
  }
}

extern "C" void kernel_launch(void* const* d_in, const int* in_sizes, int n_in,
                              void* d_out, int out_size, void* d_ws, size_t ws_size,
                              hipStream_t stream) {
  const int*   x_rows   = (const int*)  d_in[0];
  const int*   x_cols   = (const int*)  d_in[1];
  const float* x_vals   = (const float*)d_in[2];
  const int*   adj_rows = (const int*)  d_in[3];
  const int*   adj_cols = (const int*)  d_in[4];
  const float* adj_vals = (const float*)d_in[5];
  const float* W        = (const float*)d_in[6];

  const int nnz_x   = in_sizes[0];
  const int nnz_adj = in_sizes[3];

  float* out = (float*)d_out;                       // [N_NODES * 128] f32
  float* xw  = (float*)d_ws;                        // needs 25.6 MB scratch

  const int feat_elems = GCN_N_NODES * GCN_OUT_DIM; // 6.4M floats
  const int n4 = feat_elems / 4;

  // 1) Zero accumulators (d_out and d_ws are poisoned by the harness).
  {
    dim3 blk(256);
    dim3 grd((n4 + 255) / 256);
    gcn_zero2_f4<<<grd, blk, 0, stream>>>((float4*)xw, (float4*)out, n4);
  }

  // 2) xw = sparse_x @ W
  {
    dim3 blk(256);
    dim3 grd(1536);   // block-strided over 256-entry chunks
    gcn_spmm_scatter<<<grd, blk, 0, stream>>>(x_rows, x_cols, x_vals,
                                              W, xw, nnz_x);
  }

  // 3) out = sparse_adj @ xw
  {
    dim3 blk(256);
    dim3 grd(1536);
    gcn_spmm_scatter<<<grd, blk, 0, stream>>>(adj_rows, adj_cols, adj_vals,
                                              xw, out, nnz_adj);
  }

  // 4) relu in place
  {
    dim3 blk(256);
    dim3 grd((n4 + 255) / 256);
    gcn_relu_f4<<<grd, blk, 0, stream>>>((float4*)out, n4);
  }
}